// SumSquaredError_51969104281897
// MI455X (gfx1250) — compile-verified
//
#include <hip/hip_runtime.h>
#include <hip/hip_bf16.h>

typedef __attribute__((ext_vector_type(2))) float v2f;
typedef __attribute__((ext_vector_type(4))) float f4;
typedef __attribute__((ext_vector_type(8))) float v8f;

#define NTHREADS 256
#define CPB      256      // cells per block-chunk
#define FPC      30       // floats per cell (5*NB + NC)
#define NBLK     1024
#define NCH      8        // accumulator channels (6 used, padded to 8)

__device__ __forceinline__ float iou_box(const float* a, const float* b) {
    float ax1 = a[0] - a[2] * 0.5f, ay1 = a[1] - a[3] * 0.5f;
    float ax2 = a[0] + a[2] * 0.5f, ay2 = a[1] + a[3] * 0.5f;
    float bx1 = b[0] - b[2] * 0.5f, by1 = b[1] - b[3] * 0.5f;
    float bx2 = b[0] + b[2] * 0.5f, by2 = b[1] + b[3] * 0.5f;
    float iw = fmaxf(fminf(ax2, bx2) - fmaxf(ax1, bx1), 0.0f);
    float ih = fmaxf(fminf(ay2, by2) - fmaxf(ay1, by1), 0.0f);
    float inter = iw * ih;
    float aa = fmaxf(ax2 - ax1, 0.0f) * fmaxf(ay2 - ay1, 0.0f);
    float ab = fmaxf(bx2 - bx1, 0.0f) * fmaxf(by2 - by1, 0.0f);
    return inter / (aa + ab - inter + 1e-6f);
}

__global__ void __launch_bounds__(NTHREADS)
yolo_loss_partial(const float* __restrict__ gt, const float* __restrict__ pr,
                  float* __restrict__ partials, long long totalCells) {
    __shared__ __align__(16) float sGT[CPB * FPC];   // 30720 B
    __shared__ __align__(16) float sPR[CPB * FPC];   // 30720 B
    __shared__ float sRed[8 * NCH];                  // 8 waves x 8 channels

    const int tid = threadIdx.x;
    float acc[NCH];
#pragma unroll
    for (int c = 0; c < NCH; ++c) acc[c] = 0.0f;

    const long long numChunks = (totalCells + CPB - 1) / CPB;

    for (long long chunk = blockIdx.x; chunk < numChunks; chunk += (long long)gridDim.x) {
        const long long baseCell = chunk * CPB;
        long long rem = totalCells - baseCell;
        const int nc = (rem < (long long)CPB) ? (int)rem : CPB;
        const int nf = nc * FPC;

        const float* gsrc = gt + baseCell * FPC;
        const float* psrc = pr + baseCell * FPC;

        // Cooperative coalesced staging: non-temporal 128-bit streaming loads.
        const int nf4 = nf >> 2;
        const f4* g4 = (const f4*)gsrc;
        const f4* p4 = (const f4*)psrc;
        f4* sg4 = (f4*)sGT;
        f4* sp4 = (f4*)sPR;
        for (int i = tid; i < nf4; i += NTHREADS) {
            sg4[i] = __builtin_nontemporal_load(&g4[i]);
            sp4[i] = __builtin_nontemporal_load(&p4[i]);
        }
        for (int i = (nf4 << 2) + tid; i < nf; i += NTHREADS) {
            sGT[i] = gsrc[i];
            sPR[i] = psrc[i];
        }
        __syncthreads();

        if (tid < nc) {
            const float* g = &sGT[tid * FPC];
            const float* p = &sPR[tid * FPC];

            const float iou0 = iou_box(g + 0, p + 0);
            const float iou1 = iou_box(g + 5, p + 5);
            const bool hot1 = (iou1 > iou0);          // jnp.argmax: ties -> box 0
            const bool hot0 = !hot1;
            const float miou = hot1 ? iou1 : iou0;

            const bool ob0 = (g[4] == 1.0f);
            const bool ob1 = (g[9] == 1.0f);
            const bool obi = ob0;                     // obj_i = obj_box[...,0]
            const bool oo0 = ob0 && (!obi || hot0);   // one_obj_ij
            const bool oo1 = ob1 && (!obi || hot1);

            const float gc0 = obi ? (hot0 ? miou : 0.0f) : g[4];
            const float gc1 = obi ? (hot1 ? miou : 0.0f) : g[9];

            float b0 = 0.0f, b1 = 0.0f;
#pragma unroll
            for (int j = 0; j < 4; ++j) {
                float d0 = p[j] - g[j];         b0 = fmaf(d0, d0, b0);
                float d1 = p[5 + j] - g[5 + j]; b1 = fmaf(d1, d1, b1);
            }
            float e0 = p[4] - gc0; e0 *= e0;
            float e1 = p[9] - gc1; e1 *= e1;

            float cs = 0.0f;
#pragma unroll
            for (int j = 10; j < 30; ++j) {
                float d = p[j] - g[j]; cs = fmaf(d, d, cs);
            }

            acc[0] += (oo0 ? b0 : 0.0f) + (oo1 ? b1 : 0.0f);   // box SSE (obj)
            acc[1] += (oo0 ? e0 : 0.0f) + (oo1 ? e1 : 0.0f);   // obj conf SSE
            acc[2] += (oo0 ? 0.0f : e0) + (oo1 ? 0.0f : e1);   // noobj conf SSE
            acc[3] += obi ? cs : 0.0f;                          // class SSE
            acc[4] += (oo0 ? 1.0f : 0.0f) + (oo1 ? 1.0f : 0.0f);// n_obj
            acc[5] += obi ? 1.0f : 0.0f;                        // n_cls
        }
        __syncthreads();
    }

    // ---- wave32 butterfly reduction of 8 channels ----
#pragma unroll
    for (int c = 0; c < NCH; ++c) {
        float v = acc[c];
#pragma unroll
        for (int s = 16; s >= 1; s >>= 1) v += __shfl_xor(v, s, 32);
        acc[c] = v;
    }
    const int wave = tid >> 5;
    const int lane = tid & 31;
    if (lane == 0) {
#pragma unroll
        for (int c = 0; c < NCH; ++c) sRed[wave * NCH + c] = acc[c];
    }
    __syncthreads();

    // ---- cross-wave reduce via V_WMMA_F32_16X16X4_F32 (D = A x ones + 0) ----
    // A layout (ISA 16x4 f32): lane l, vgpr v -> M = l%16, K = 2*(l/16)+v.
    // We place A[m][k] = sRed[wave(m,k)][chan(m)] with wave = (m>>3)*4 + k,
    // chan = m&7, so rowsum(m) = sum over 4 waves of channel chan(m).
    // D[c][n] = waves 0..3, D[c+8][n] = waves 4..7  =>  fold with shfl_xor(16).
    if (tid < 32) {
        const int m = tid & 15;
        const int kbase = (tid >> 4) << 1;     // 0 or 2
        const int chan = m & 7;
        const int wbase = (m >> 3) << 2;       // 0 or 4
        v2f a;
        a.x = sRed[(wbase + kbase + 0) * NCH + chan];
        a.y = sRed[(wbase + kbase + 1) * NCH + chan];
        v2f bones;
        bones.x = 1.0f;
        bones.y = 1.0f;
        v8f c8 = {};
        v8f d = __builtin_amdgcn_wmma_f32_16x16x4_f32(
            false, a, false, bones, (short)0, c8, false, false);
#pragma unroll
        for (int c = 0; c < NCH; ++c) {
            float v = d[c] + __shfl_xor(d[c], 16, 32);
            if (tid == 0) partials[(long long)blockIdx.x * NCH + c] = v;
        }
    }
}

__global__ void yolo_loss_final(const float* __restrict__ partials, int nblk,
                                float* __restrict__ out, long long totalCells) {
    float acc[NCH];
#pragma unroll
    for (int c = 0; c < NCH; ++c) acc[c] = 0.0f;

    for (int r = threadIdx.x; r < nblk; r += 32) {
#pragma unroll
        for (int c = 0; c < NCH; ++c) acc[c] += partials[r * NCH + c];
    }
#pragma unroll
    for (int c = 0; c < NCH; ++c) {
#pragma unroll
        for (int s = 16; s >= 1; s >>= 1) acc[c] += __shfl_xor(acc[c], s, 32);
    }
    if (threadIdx.x == 0) {
        const float sbox = acc[0], sobj = acc[1], snoobj = acc[2], scls = acc[3];
        const float nobj = acc[4], ncls = acc[5];
        const float totalBoxes = (float)(2 * totalCells);
        const float Nobj   = fmaxf(nobj, 1.0f);
        const float Nnoobj = fmaxf(totalBoxes - nobj, 1.0f);
        const float Ncls   = fmaxf(ncls, 1.0f);
        out[0] = 5.0f * sbox / (Nobj * 4.0f);                 // LAMBDA_COORD * box_loss
        out[1] = 0.5f * snoobj / Nnoobj + sobj / Nobj;        // LAMBDA_NOOBJ*noobj + obj
        out[2] = scls / (Ncls * 20.0f);                       // cls_loss
    }
}

extern "C" void kernel_launch(void* const* d_in, const int* in_sizes, int n_in,
                              void* d_out, int out_size, void* d_ws, size_t ws_size,
                              hipStream_t stream) {
    const float* gt = (const float*)d_in[0];
    const float* pr = (const float*)d_in[1];
    float* out = (float*)d_out;
    float* partials = (float*)d_ws;              // NBLK * 8 floats = 32 KB

    const long long totalCells = (long long)in_sizes[0] / FPC;

    yolo_loss_partial<<<NBLK, NTHREADS, 0, stream>>>(gt, pr, partials, totalCells);
    yolo_loss_final<<<1, 32, 0, stream>>>(partials, NBLK, out, totalCells);
}